// HeteroHypergraphTransformerConv_30408368456296
// MI455X (gfx1250) — compile-verified
//
#include <hip/hip_runtime.h>
#include <math.h>

#define N_NODES 100000
#define E_ENT   1000000
#define NUM_HE  20000
#define DIM     128
#define HEADS   4
#define ROWS_PB 32              // rows per block in the WMMA GEMM (2 x 16-row tiles)

typedef float v2f __attribute__((ext_vector_type(2)));
typedef float v8f __attribute__((ext_vector_type(8)));

__device__ __forceinline__ void atomic_add_f32(float* p, float v) {
    __hip_atomic_fetch_add(p, v, __ATOMIC_RELAXED, __HIP_MEMORY_SCOPE_AGENT);
}

// order-preserving float->uint encoding so u32 atomicMax == float max
__device__ __forceinline__ unsigned enc_f32(float f) {
    unsigned u = __float_as_uint(f);
    return (u & 0x80000000u) ? ~u : (u | 0x80000000u);
}
__device__ __forceinline__ float dec_f32(unsigned u) {
    return __uint_as_float((u & 0x80000000u) ? (u & 0x7fffffffu) : ~u);
}

// ---------------------------------------------------------------------------
// out[M,128] = A[M,128] @ W[128,128] + bias   (fp32 WMMA 16x16x4, wave32)
// Block: 256 threads = 8 waves, 32 rows. Wave w owns 16-col tile w and BOTH
// 16-row tiles: two independent accumulator chains hide the WMMA->WMMA RAW
// hazard, and each B fragment is reused by two WMMAs (halves B-load traffic).
// A 32x128 tile staged in LDS (16 KB), shared by all 8 waves.
// ---------------------------------------------------------------------------
__global__ __launch_bounds__(256)
void wmma_gemm_bias_f32(const float* __restrict__ A,
                        const float* __restrict__ W,
                        const float* __restrict__ bias,
                        float* __restrict__ out) {
    __shared__ float As[ROWS_PB * DIM];       // 16 KB
    const int t = threadIdx.x;
    const long base = (long)blockIdx.x * (ROWS_PB * DIM);
    const float4* __restrict__ Ag = (const float4*)(A + base);
    float4* As4 = (float4*)As;
#pragma unroll
    for (int i = 0; i < 4; ++i)               // 1024 float4 / 256 threads
        As4[t + i * 256] = Ag[t + i * 256];
    __syncthreads();

    const int lane  = t & 31;
    const int wv    = t >> 5;                 // 0..7 column tile
    const int ntile = wv * 16;
    const int mrow  = lane & 15;              // A: M for this lane
    const int koff  = (lane >> 4) * 2;        // A/B: K sub-offset per half-wave
    const int ncol  = lane & 15;              // B/C/D: N for this lane

    v8f c0 = {0.f, 0.f, 0.f, 0.f, 0.f, 0.f, 0.f, 0.f};
    v8f c1 = {0.f, 0.f, 0.f, 0.f, 0.f, 0.f, 0.f, 0.f};
    for (int k = 0; k < DIM; k += 4) {
        v2f a0, a1, b;
        a0.x = As[mrow * DIM + k + koff];
        a0.y = As[mrow * DIM + k + koff + 1];
        a1.x = As[(mrow + 16) * DIM + k + koff];
        a1.y = As[(mrow + 16) * DIM + k + koff + 1];
        b.x  = W[(k + koff) * DIM + ntile + ncol];
        b.y  = W[(k + koff + 1) * DIM + ntile + ncol];
        c0 = __builtin_amdgcn_wmma_f32_16x16x4_f32(false, a0, false, b,
                                                   (short)0, c0, false, false);
        c1 = __builtin_amdgcn_wmma_f32_16x16x4_f32(false, a1, false, b,
                                                   (short)0, c1, false, false);
    }
    const float bv = bias[ntile + ncol];
    const long row0 = (long)blockIdx.x * ROWS_PB + ((lane >> 4) << 3);
#pragma unroll
    for (int i = 0; i < 8; ++i) {
        out[(row0 + i) * DIM + ntile + ncol]      = c0[i] + bv;
        out[(row0 + 16 + i) * DIM + ntile + ncol] = c1[i] + bv;
    }
}

// he_feat[he] += h_node[node]  (one wave per incidence entry, lane = float4)
__global__ __launch_bounds__(256)
void scatter_he_kernel(const float* __restrict__ h_node,
                       const int* __restrict__ node_idx,
                       const int* __restrict__ he_idx,
                       float* __restrict__ he_feat) {
    const int lane = threadIdx.x & 31;
    const int e = blockIdx.x * 8 + (threadIdx.x >> 5);
    if (e >= E_ENT) return;
    const int node = node_idx[e];
    const int he   = he_idx[e];
    const float4 v = *(const float4*)(h_node + (long)node * DIM + lane * 4);
    float* dst = he_feat + (long)he * DIM + lane * 4;
    atomic_add_f32(dst + 0, v.x);
    atomic_add_f32(dst + 1, v.y);
    atomic_add_f32(dst + 2, v.z);
    atomic_add_f32(dst + 3, v.w);
}

// alpha[e,h] = dot(q[node,h,:], he_feat[he,h,:]) / sqrt(32); per-node max
__global__ __launch_bounds__(256)
void alpha_max_kernel(const float* __restrict__ q,
                      const float* __restrict__ he_feat,
                      const int* __restrict__ node_idx,
                      const int* __restrict__ he_idx,
                      float* __restrict__ alpha,
                      unsigned* __restrict__ mmax) {
    const int lane = threadIdx.x & 31;
    const int e = blockIdx.x * 8 + (threadIdx.x >> 5);
    if (e >= E_ENT) return;
    const int node = node_idx[e];
    const int he   = he_idx[e];
    const float4 qv = *(const float4*)(q + (long)node * DIM + lane * 4);
    const float4 hv = *(const float4*)(he_feat + (long)he * DIM + lane * 4);
    float p = qv.x * hv.x + qv.y * hv.y + qv.z * hv.z + qv.w * hv.w;
    // reduce within each 8-lane head group (8 lanes x float4 = dk = 32)
    p += __shfl_xor(p, 1, 8);
    p += __shfl_xor(p, 2, 8);
    p += __shfl_xor(p, 4, 8);
    const float a = p * 0.17677669529663689f; // 1/sqrt(32)
    if ((lane & 7) == 0) {
        const int h = lane >> 3;
        alpha[(long)e * HEADS + h] = a;
        atomicMax(mmax + (long)node * HEADS + h, enc_f32(a));
    }
}

// ssum[node,h] += exp(alpha[e,h] - m[node,h])
__global__ __launch_bounds__(256)
void alpha_sum_kernel(const float* __restrict__ alpha,
                      const int* __restrict__ node_idx,
                      const unsigned* __restrict__ mmax,
                      float* __restrict__ ssum) {
    const long idx = (long)blockIdx.x * 256 + threadIdx.x;
    if (idx >= (long)E_ENT * HEADS) return;
    const int e = (int)(idx >> 2);
    const int h = (int)(idx & 3);
    const int node = node_idx[e];
    const float a  = alpha[idx];
    const float mv = dec_f32(mmax[node * HEADS + h]);
    atomic_add_f32(ssum + node * HEADS + h, expf(a - mv));
}

// acc[node] += he_feat[he] * softmax_weight   (acc pre-initialized with x)
__global__ __launch_bounds__(256)
void scatter_out_kernel(const float* __restrict__ he_feat,
                        const int* __restrict__ node_idx,
                        const int* __restrict__ he_idx,
                        const float* __restrict__ alpha,
                        const unsigned* __restrict__ mmax,
                        const float* __restrict__ ssum,
                        float* __restrict__ acc) {
    const int lane = threadIdx.x & 31;
    const int e = blockIdx.x * 8 + (threadIdx.x >> 5);
    if (e >= E_ENT) return;
    const int node = node_idx[e];
    const int he   = he_idx[e];
    const int h    = lane >> 3;
    const float a  = alpha[(long)e * HEADS + h];
    const float mv = dec_f32(mmax[node * HEADS + h]);
    const float sv = ssum[node * HEADS + h];
    const float w  = expf(a - mv) / (sv + 1e-16f);
    const float4 hv = *(const float4*)(he_feat + (long)he * DIM + lane * 4);
    float* dst = acc + (long)node * DIM + lane * 4;
    atomic_add_f32(dst + 0, hv.x * w);
    atomic_add_f32(dst + 1, hv.y * w);
    atomic_add_f32(dst + 2, hv.z * w);
    atomic_add_f32(dst + 3, hv.w * w);
}

__global__ void init_m_kernel(unsigned* __restrict__ m, int n) {
    const int i = blockIdx.x * blockDim.x + threadIdx.x;
    if (i < n) m[i] = 0x007fffffu;  // enc_f32(-inf)
}

extern "C" void kernel_launch(void* const* d_in, const int* in_sizes, int n_in,
                              void* d_out, int out_size, void* d_ws, size_t ws_size,
                              hipStream_t stream) {
    const float* x        = (const float*)d_in[0];
    const int*   node_idx = (const int*)d_in[1];
    const int*   he_idx   = (const int*)d_in[2];
    const float* Wk = (const float*)d_in[3];
    const float* bk = (const float*)d_in[4];
    const float* Wq = (const float*)d_in[5];
    const float* bq = (const float*)d_in[6];
    const float* Wa = (const float*)d_in[7];
    const float* ba = (const float*)d_in[8];
    float* out = (float*)d_out;

    // workspace layout (fp32 elements)
    float*    h_node  = (float*)d_ws;                                    // N*128
    float*    q       = h_node  + (size_t)N_NODES * DIM;                 // N*128
    float*    he_feat = q       + (size_t)N_NODES * DIM;                 // HE*128
    float*    alpha   = he_feat + (size_t)NUM_HE * DIM;                  // E*4
    unsigned* mmax    = (unsigned*)(alpha + (size_t)E_ENT * HEADS);      // N*4
    float*    ssum    = (float*)(mmax + (size_t)N_NODES * HEADS);        // N*4
    float*    acc     = ssum + (size_t)N_NODES * HEADS;                  // N*128

    hipMemsetAsync(he_feat, 0, (size_t)NUM_HE * DIM * sizeof(float), stream);
    hipMemsetAsync(ssum, 0, (size_t)N_NODES * HEADS * sizeof(float), stream);
    hipMemcpyAsync(acc, x, (size_t)N_NODES * DIM * sizeof(float),
                   hipMemcpyDeviceToDevice, stream);  // residual pre-load
    init_m_kernel<<<(N_NODES * HEADS + 255) / 256, 256, 0, stream>>>(
        mmax, N_NODES * HEADS);

    // K and Q projections (100000 = 3125 * 32 rows)
    wmma_gemm_bias_f32<<<N_NODES / ROWS_PB, 256, 0, stream>>>(x, Wk, bk, h_node);
    wmma_gemm_bias_f32<<<N_NODES / ROWS_PB, 256, 0, stream>>>(x, Wq, bq, q);

    scatter_he_kernel<<<E_ENT / 8, 256, 0, stream>>>(h_node, node_idx, he_idx,
                                                     he_feat);
    alpha_max_kernel<<<E_ENT / 8, 256, 0, stream>>>(q, he_feat, node_idx,
                                                    he_idx, alpha, mmax);
    alpha_sum_kernel<<<(E_ENT * HEADS + 255) / 256, 256, 0, stream>>>(
        alpha, node_idx, mmax, ssum);
    scatter_out_kernel<<<E_ENT / 8, 256, 0, stream>>>(he_feat, node_idx, he_idx,
                                                      alpha, mmax, ssum, acc);

    // out = (node_out + x) @ Wa + ba
    wmma_gemm_bias_f32<<<N_NODES / ROWS_PB, 256, 0, stream>>>(acc, Wa, ba, out);
}